// ProperYOLOLoss_67250597921095
// MI455X (gfx1250) — compile-verified
//
#include <hip/hip_runtime.h>
#include <hip/hip_bf16.h>

// YOLO loss on MI455X (gfx1250, wave32).
// Roofline: ~268 MB unique traffic -> ~11.5us @ 23.3 TB/s; VALU cost is the
// same order, so per-batch threads hoist the GT corner/area math (computed
// once, reused by all 3 anchors) and remove all redundant gt loads.
// Reduction: V_WMMA_F32_16X16X4_F32 with ones-B collapses a wave's 32
// partials on the matrix unit, then LDS across waves, then a deterministic
// single-block second pass scaled by 1/B.

typedef __attribute__((ext_vector_type(2))) float v2f;
typedef __attribute__((ext_vector_type(8))) float v8f;

#define LAMBDA_COORD 5.0f

__device__ __forceinline__ float fast_sigmoid(float x) {
    return 1.0f / (1.0f + __expf(-x));
}

// Wave32 sum-reduction via WMMA: A[16x4] holds the 32 lane partials
// (lane L -> A[M=L%16, K = L<16 ? 0 : 2], K=1/3 zeroed), B = ones.
// Summing D's 8 VGPRs per lane leaves lanes 0-15 with one half-sum and
// lanes 16-31 with the other; one cross-lane add finishes the wave.
__device__ __forceinline__ float wave_reduce_sum_wmma(float v) {
    v2f a; a[0] = v;    a[1] = 0.0f;
    v2f b; b[0] = 1.0f; b[1] = 1.0f;
    v8f c = {};
    v8f d = __builtin_amdgcn_wmma_f32_16x16x4_f32(
        /*neg_a=*/false, a, /*neg_b=*/false, b,
        /*c_mod=*/(short)0, c, /*reuse_a=*/false, /*reuse_b=*/false);
    float s = ((d[0] + d[1]) + (d[2] + d[3])) + ((d[4] + d[5]) + (d[6] + d[7]));
    return __shfl(s, 0, 32) + __shfl(s, 16, 32);
}

// Returns full block sum in thread 0 (undefined elsewhere).
__device__ __forceinline__ float block_reduce_sum(float v, float* lds8) {
    float w = wave_reduce_sum_wmma(v);
    const int lane = threadIdx.x & 31;
    const int wave = threadIdx.x >> 5;
    if (lane == 0) lds8[wave] = w;
    __syncthreads();
    float s = 0.0f;
    if (threadIdx.x == 0) {
        const int nw = blockDim.x >> 5;
        #pragma unroll
        for (int j = 0; j < 8; ++j)
            if (j < nw) s += lds8[j];
    }
    return s;
}

__global__ void __launch_bounds__(256)
yolo_partial_kernel(const float* __restrict__ pred,
                    const float* __restrict__ gtb,
                    const int*   __restrict__ gtc,
                    float*       __restrict__ ws) {
    const int b = blockIdx.x * 256 + threadIdx.x;   // one batch per thread

    // ---- GT prep, once per batch: 8 b128 loads + corners/areas in regs ----
    const float4* g4 = reinterpret_cast<const float4*>(gtb) + 8 * (size_t)b;
    float gx1[8], gx2[8], gy1[8], gy2[8], garea[8];
    #pragma unroll
    for (int n = 0; n < 8; ++n) {
        const float4 g = g4[n];
        const float gwh = g.z * 0.5f, ghh = g.w * 0.5f;
        gx1[n] = g.x - gwh;  gx2[n] = g.x + gwh;
        gy1[n] = g.y - ghh;  gy2[n] = g.y + ghh;
        garea[n] = (gx2[n] - gx1[n]) * (gy2[n] - gy1[n]);
    }

    const float4* p4 = reinterpret_cast<const float4*>(pred) + 6 * (size_t)b;
    const int*    cb = gtc + 8 * (size_t)b;

    float partial = 0.0f;

    #pragma unroll
    for (int a = 0; a < 3; ++a) {
        const float4 pA = p4[2 * a + 0];
        const float4 pB = p4[2 * a + 1];

        const float px    = fast_sigmoid(pA.x);
        const float py    = fast_sigmoid(pA.y);
        const float pw    = pA.z;
        const float ph    = pA.w;
        const float pconf = fast_sigmoid(pB.x);
        const float l0 = pB.y, l1 = pB.z, l2 = pB.w;

        const float px1 = px - pw * 0.5f, px2 = px + pw * 0.5f;
        const float py1 = py - ph * 0.5f, py2 = py + ph * 0.5f;
        const float pred_area = (px2 - px1) * (py2 - py1);

        float best_iou = -1.0f;
        int   best_idx = 0;
        #pragma unroll
        for (int n = 0; n < 8; ++n) {
            const float iw = fmaxf(fminf(px2, gx2[n]) - fmaxf(px1, gx1[n]), 0.0f);
            const float ih = fmaxf(fminf(py2, gy2[n]) - fmaxf(py1, gy1[n]), 0.0f);
            const float inter = iw * ih;
            const float iou = inter / (pred_area + garea[n] - inter + 1e-6f);
            if (iou > best_iou) { best_iou = iou; best_idx = n; }  // first-max
        }

        const bool matched = best_iou > 0.5f;
        const float4 mb = g4[best_idx];      // L0 hit, per-lane indexed
        const int  mcls = cb[best_idx];

        // coord loss
        const float dx = px - mb.x, dy = py - mb.y, dw = pw - mb.z, dh = ph - mb.w;
        const float coord = dx * dx + dy * dy + dw * dw + dh * dh;

        // confidence BCE with clamp at -100
        const float logp   = fmaxf(__logf(pconf), -100.0f);
        const float log1mp = fmaxf(__logf(1.0f - pconf), -100.0f);
        const float conf_term = matched ? -logp : -log1mp;

        // 3-way log-softmax NLL
        const float m   = fmaxf(l0, fmaxf(l1, l2));
        const float lse = m + __logf(__expf(l0 - m) + __expf(l1 - m) + __expf(l2 - m));
        const float lsel = (mcls == 0) ? l0 : ((mcls == 1) ? l1 : l2);
        const float nll  = lse - lsel;

        partial += (matched ? (LAMBDA_COORD * coord + nll) : 0.0f) + conf_term;
    }

    __shared__ float lds8[8];
    const float bsum = block_reduce_sum(partial, lds8);
    if (threadIdx.x == 0) ws[blockIdx.x] = bsum;
}

__global__ void __launch_bounds__(256)
yolo_reduce_kernel(const float* __restrict__ ws,
                   float*       __restrict__ out,
                   int nblocks, float scale) {
    float s = 0.0f;
    // nblocks (4096) is a multiple of 256 -> uniform trip count,
    // EXEC stays all-ones into the WMMA reduction.
    for (int j = threadIdx.x; j < nblocks; j += 256) s += ws[j];
    __shared__ float lds8[8];
    const float tot = block_reduce_sum(s, lds8);
    if (threadIdx.x == 0) out[0] = tot * scale;
}

extern "C" void kernel_launch(void* const* d_in, const int* in_sizes, int n_in,
                              void* d_out, int out_size, void* d_ws, size_t ws_size,
                              hipStream_t stream) {
    const float* pred = (const float*)d_in[0];   // (B, 3, 8) f32
    const float* gtb  = (const float*)d_in[1];   // (B, 8, 4) f32
    const int*   gtc  = (const int*)d_in[2];     // (B, 8)    i32
    float* out = (float*)d_out;
    float* ws  = (float*)d_ws;

    const int batch   = in_sizes[0] / 24;        // B = 1,048,576
    const int nblocks = batch / 256;             // 4096, exact

    yolo_partial_kernel<<<nblocks, 256, 0, stream>>>(pred, gtb, gtc, ws);
    yolo_reduce_kernel<<<1, 256, 0, stream>>>(ws, out, nblocks,
                                              1.0f / (float)batch);
}